// GraphConvolution_26774826123627
// MI455X (gfx1250) — compile-verified
//
#include <hip/hip_runtime.h>

typedef __attribute__((ext_vector_type(2))) float v2f;
typedef __attribute__((ext_vector_type(8))) float v8f;

#define BATCH 4
#define NNODE 50000
#define DINF  64
#define DOUTF 64
#define NROWS (BATCH * NNODE)          // 200000
#define FEAT  (BATCH * DOUTF)          // 256 (SpMM row width)
#define EPSBN 1e-5f

// ---------------------------------------------------------------- zero utils
__global__ void zero_i32(int* __restrict__ p, int n) {
    int i = blockIdx.x * blockDim.x + threadIdx.x;
    if (i < n) p[i] = 0;
}

__global__ void zero_f1(float* __restrict__ p, int n) {
    int i = blockIdx.x * blockDim.x + threadIdx.x;
    if (i < n) p[i] = 0.f;
}

// ------------------------------------------------- GEMM (WMMA f32) + BN stats
// Block: 256 threads = 8 waves. Block tile: 32 rows x 64 cols of X = H @ W.
// Wave w -> row-subtile (w>>2)*16, col-subtile (w&3)*16.
// Writes X directly in Xn layout: Xn[n*256 + b*64 + d], r = b*N + n.
__global__ __launch_bounds__(256)
void gemm_bn_stats(const float* __restrict__ H, const float* __restrict__ W,
                   float* __restrict__ Y, float* __restrict__ sums,
                   float* __restrict__ sqs) {
    __shared__ float hs[32 * 64];   // H tile, 8 KB
    __shared__ float wl[64 * 64];   // full W, 16 KB

    const int tid = threadIdx.x;
    const long gbase = (long)blockIdx.x * 32 * 64;

    // Coalesced stage of H tile (2048 f32) and W (4096 f32) into LDS.
    #pragma unroll
    for (int j = 0; j < 2; ++j) {
        int lin = j * 1024 + tid * 4;
        *(float4*)&hs[lin] = *(const float4*)&H[gbase + lin];
    }
    #pragma unroll
    for (int j = 0; j < 4; ++j) {
        int lin = j * 1024 + tid * 4;
        *(float4*)&wl[lin] = *(const float4*)&W[lin];
    }
    __syncthreads();

    const int lane  = tid & 31;
    const int wave  = tid >> 5;
    const int lrow0 = (wave >> 2) * 16;        // 0 or 16 within block tile
    const int n0    = (wave & 3) * 16;         // column subtile
    const int m     = lane & 15;
    const int kk    = (lane >> 4) << 1;        // 0 (lanes 0-15) or 2 (16-31)
    const int col   = n0 + m;

    // K-loop: 16 x v_wmma_f32_16x16x4_f32 accumulating a 16x16 f32 tile.
    v8f c = {0.f, 0.f, 0.f, 0.f, 0.f, 0.f, 0.f, 0.f};
    #pragma unroll
    for (int k0 = 0; k0 < DINF; k0 += 4) {
        // A fragment: 16x4 f32, ISA layout (VGPR0: K=kk, VGPR1: K=kk+1)
        v2f a = *(const v2f*)&hs[(lrow0 + m) * 64 + k0 + kk];
        // B fragment: 4x16 f32, mirrored layout
        v2f b;
        b.x = wl[(k0 + kk) * 64 + col];
        b.y = wl[(k0 + kk + 1) * 64 + col];
        c = __builtin_amdgcn_wmma_f32_16x16x4_f32(
                false, a, false, b, (short)0, c, false, false);
    }

    // C/D layout: VGPR i holds M = i (lanes 0-15) or M = 8+i (lanes 16-31).
    const int row0  = blockIdx.x * 32 + lrow0;
    const int mbase = (lane >> 4) * 8;
    float s = 0.f, q = 0.f;
    #pragma unroll
    for (int i = 0; i < 8; ++i) {
        int r  = row0 + mbase + i;      // flat row in [B*N]
        int bb = r / NNODE;
        int nn = r - bb * NNODE;
        float v = c[i];
        Y[(long)nn * FEAT + bb * DOUTF + col] = v;   // Xn layout
        s += v;
        q += v * v;
    }
    // lanes L and L+16 hold the same feature 'col' -> fold halves, 16 atomics.
    s += __shfl_down(s, 16, 32);
    q += __shfl_down(q, 16, 32);
    if (lane < 16) {
        atomicAdd(&sums[col], s);
        atomicAdd(&sqs[col], q);
    }
}

// -------------------------------------------------------------- BN finalize
__global__ void bn_finalize(const float* __restrict__ sums,
                            const float* __restrict__ sqs,
                            const float* __restrict__ gamma,
                            const float* __restrict__ beta,
                            float* __restrict__ scale,
                            float* __restrict__ shift) {
    int d = threadIdx.x;
    if (d < DOUTF) {
        const float inv = 1.0f / (float)NROWS;
        float mean = sums[d] * inv;
        float var  = sqs[d] * inv - mean * mean;
        float sc   = gamma[d] * rsqrtf(var + EPSBN);
        scale[d] = sc;
        shift[d] = beta[d] - mean * sc;
    }
}

// ------------------------------------------------------- BN apply + ReLU
__global__ __launch_bounds__(256)
void bn_relu(float* __restrict__ Y, const float* __restrict__ scale,
             const float* __restrict__ shift) {
    long i = (long)blockIdx.x * blockDim.x + threadIdx.x;   // one float4 each
    long base = i * 4;
    int d0 = (int)(base & 63);      // feature = (pos % 256) % 64
    float4 v = ((float4*)Y)[i];
    v.x = fmaxf(0.f, v.x * scale[d0 + 0] + shift[d0 + 0]);
    v.y = fmaxf(0.f, v.y * scale[d0 + 1] + shift[d0 + 1]);
    v.z = fmaxf(0.f, v.z * scale[d0 + 2] + shift[d0 + 2]);
    v.w = fmaxf(0.f, v.w * scale[d0 + 3] + shift[d0 + 3]);
    ((float4*)Y)[i] = v;
}

// --------------------------------------------------------- CSR build: count
__global__ void csr_count(const int* __restrict__ rows, int* __restrict__ counts,
                          int nedges) {
    int e = blockIdx.x * blockDim.x + threadIdx.x;
    if (e < nedges) atomicAdd(&counts[rows[e]], 1);
}

// ---------------------------------------------------------- CSR build: scan
// Single 1024-thread block: chunked Hillis-Steele exclusive scan with carry.
__global__ __launch_bounds__(1024)
void csr_scan(const int* __restrict__ counts, int* __restrict__ offsets,
              int* __restrict__ cursor) {
    __shared__ int buf[1024];
    __shared__ int carry;
    if (threadIdx.x == 0) carry = 0;
    __syncthreads();
    for (int base = 0; base < NNODE; base += 1024) {
        int i = base + threadIdx.x;
        int v = (i < NNODE) ? counts[i] : 0;
        buf[threadIdx.x] = v;
        __syncthreads();
        #pragma unroll
        for (int off = 1; off < 1024; off <<= 1) {
            int t = (threadIdx.x >= off) ? buf[threadIdx.x - off] : 0;
            __syncthreads();
            buf[threadIdx.x] += t;
            __syncthreads();
        }
        int excl = buf[threadIdx.x] - v + carry;   // reads old carry
        if (i < NNODE) { offsets[i] = excl; cursor[i] = excl; }
        __syncthreads();
        if (threadIdx.x == 1023) carry += buf[1023];
        __syncthreads();
    }
    if (threadIdx.x == 0) offsets[NNODE] = carry;
}

// ------------------------------------------------------- CSR build: scatter
__global__ void csr_scatter(const int* __restrict__ rows,
                            const int* __restrict__ cols,
                            const float* __restrict__ vals,
                            int* __restrict__ cursor,
                            int* __restrict__ ecol, float* __restrict__ eval,
                            int nedges) {
    int e = blockIdx.x * blockDim.x + threadIdx.x;
    if (e < nedges) {
        int p = atomicAdd(&cursor[rows[e]], 1);
        ecol[p] = cols[e];
        eval[p] = vals[e];
    }
}

// ------------------------------------------------------------------- SpMM
// One wave per output row: register accumulation of the full 256-wide row
// (8 f32 per lane), cooperative edge-metadata loads + shfl broadcast,
// zero atomics, plain b128 stores (covers d_out entirely -> no zero pass).
__global__ __launch_bounds__(256)
void spmm_csr(const int* __restrict__ offsets, const int* __restrict__ ecol,
              const float* __restrict__ eval, const float* __restrict__ Y,
              float* __restrict__ out) {
    const int lane = threadIdx.x & 31;
    const int wave = threadIdx.x >> 5;
    const int row  = blockIdx.x * 8 + wave;
    if (row >= NNODE) return;

    const int beg = offsets[row];
    const int end = offsets[row + 1];

    float acc[8] = {0.f, 0.f, 0.f, 0.f, 0.f, 0.f, 0.f, 0.f};
    const float* ybase = Y + lane * 8;

    for (int p = beg; p < end; p += 32) {
        const int cnt = min(32, end - p);
        int   myc = 0;
        float myv = 0.f;
        if (lane < cnt) { myc = ecol[p + lane]; myv = eval[p + lane]; }
        for (int j = 0; j < cnt; ++j) {
            int   cj = __shfl(myc, j, 32);
            float vj = __shfl(myv, j, 32);
            const float4* src = (const float4*)(ybase + (long)cj * FEAT);
            float4 x0 = src[0];
            float4 x1 = src[1];
            acc[0] += vj * x0.x;  acc[1] += vj * x0.y;
            acc[2] += vj * x0.z;  acc[3] += vj * x0.w;
            acc[4] += vj * x1.x;  acc[5] += vj * x1.y;
            acc[6] += vj * x1.z;  acc[7] += vj * x1.w;
        }
    }

    // lane covers features lane*8 .. lane*8+7 -> b = lane>>3, d0 = (lane&7)*8
    const int b  = lane >> 3;
    const int d0 = (lane & 7) * 8;
    float4 o0 = make_float4(acc[0], acc[1], acc[2], acc[3]);
    float4 o1 = make_float4(acc[4], acc[5], acc[6], acc[7]);
    float4* dst = (float4*)(out + (long)b * NNODE * DOUTF + (long)row * DOUTF + d0);
    dst[0] = o0;
    dst[1] = o1;
}

// ------------------------------------------------------------------ launch
extern "C" void kernel_launch(void* const* d_in, const int* in_sizes, int n_in,
                              void* d_out, int out_size, void* d_ws, size_t ws_size,
                              hipStream_t stream) {
    const float* H      = (const float*)d_in[0];
    const float* W      = (const float*)d_in[1];
    const float* gamma  = (const float*)d_in[2];
    const float* beta   = (const float*)d_in[3];
    const int*   A_rows = (const int*)d_in[4];
    const int*   A_cols = (const int*)d_in[5];
    const float* A_vals = (const float*)d_in[6];
    float* out = (float*)d_out;
    const int nedges = in_sizes[4];

    // Workspace layout (all 4-byte elements; regions padded to 16B multiples)
    float* Y      = (float*)d_ws;                        // [N*256]  51.2 MB
    float* sums   = Y + (size_t)NROWS * DOUTF;           // [64]
    float* sqs    = sums + DOUTF;                        // [64]
    float* scale  = sqs + DOUTF;                         // [64]
    float* shift  = scale + DOUTF;                       // [64]
    int*   counts = (int*)(shift + DOUTF);               // [N]
    int*   offs   = counts + NNODE;                      // [N+1] (pad to +4)
    int*   cursor = offs + NNODE + 4;                    // [N]
    int*   ecol   = cursor + NNODE;                      // [E]
    float* eval_  = (float*)(ecol + nedges);             // [E]

    // 1) zero accumulators (every call; graph-replay safe)
    zero_i32<<<(NNODE + 255) / 256, 256, 0, stream>>>(counts, NNODE);
    zero_f1<<<1, 128, 0, stream>>>(sums, 128);           // sums + sqs

    // 2) CSR build: count -> scan -> scatter  (~1.6M int atomics total)
    csr_count<<<(nedges + 255) / 256, 256, 0, stream>>>(A_rows, counts, nedges);
    csr_scan<<<1, 1024, 0, stream>>>(counts, offs, cursor);
    csr_scatter<<<(nedges + 255) / 256, 256, 0, stream>>>(
        A_rows, A_cols, A_vals, cursor, ecol, eval_, nedges);

    // 3) WMMA GEMM + BN statistics (200000/32 = 6250 blocks)
    gemm_bn_stats<<<NROWS / 32, 256, 0, stream>>>(H, W, Y, sums, sqs);

    // 4) finalize BN scale/shift
    bn_finalize<<<1, 64, 0, stream>>>(sums, sqs, gamma, beta, scale, shift);

    // 5) BN + ReLU elementwise (12.8M floats / 4 / 256 = 12500 blocks)
    bn_relu<<<(NROWS * DOUTF) / 4 / 256, 256, 0, stream>>>(Y, scale, shift);

    // 6) SpMM: one wave per row, register accumulation, no atomics
    spmm_csr<<<(NNODE + 7) / 8, 256, 0, stream>>>(offs, ecol, eval_, Y, out);
}